// GLSimCrop_1159641170176
// MI455X (gfx1250) — compile-verified
//
#include <hip/hip_runtime.h>
#include <hip/hip_bf16.h>

typedef __attribute__((ext_vector_type(2))) float v2f;
typedef __attribute__((ext_vector_type(8))) float v8f;

#define DIM     768
#define ROWS    785
#define TOKENS  784
#define IMG     448
#define FH      28      // int(sqrt(785))
#define PATCH   16
#define NTOP    8

// ---------------------------------------------------------------------------
// Kernel 1: per-batch cosine similarity (WMMA f32 16x16x4), top-8, crop box.
// One workgroup (256 threads = 8 waves) per batch.
// ---------------------------------------------------------------------------
__global__ void __launch_bounds__(256)
gl_sim_topk_kernel(const float* __restrict__ x, int4* __restrict__ boxes)
{
    __shared__ float s_dot[TOKENS];   // dots, then distances
    __shared__ float s_n2[TOKENS];    // per-token squared norms
    __shared__ float s_red[256];
    __shared__ int   s_redi[256];
    __shared__ float s_gn;

    const int b   = blockIdx.x;
    const int tid = threadIdx.x;
    const float* __restrict__ xb = x + (size_t)b * ROWS * DIM;
    const float* __restrict__ g  = xb;          // row 0 = global token
    const float* __restrict__ l  = xb + DIM;    // rows 1..784 = local tokens

    // ---- ||g|| (block reduction) ----
    float gp = 0.0f;
    for (int i = tid; i < DIM; i += 256) { float v = g[i]; gp = fmaf(v, v, gp); }
    s_red[tid] = gp;
    __syncthreads();
    for (int s = 128; s > 0; s >>= 1) {
        if (tid < s) s_red[tid] += s_red[tid + s];
        __syncthreads();
    }
    if (tid == 0) s_gn = fmaxf(sqrtf(s_red[0]), 1e-8f);
    __syncthreads();

    // ---- dots + token norms via WMMA f32 16x16x4 ----
    // A (16x4): lane m (0-15): K = k,k+1 in VGPR0/1 ; lane m+16: K = k+2,k+3.
    // B (4x16): every column is g, so per-lane indexing mirrors A.
    const int wave = tid >> 5;
    const int lane = tid & 31;
    const int m    = lane & 15;
    const int half = lane >> 4;

    for (int tile = wave; tile < TOKENS / 16; tile += 8) {
        const float* __restrict__ lrow = l + (size_t)(tile * 16 + m) * DIM;
        v8f acc0 = {0.f,0.f,0.f,0.f,0.f,0.f,0.f,0.f};
        v8f acc1 = {0.f,0.f,0.f,0.f,0.f,0.f,0.f,0.f};
        float n2 = 0.0f;
        #pragma unroll 2
        for (int k = 0; k < DIM; k += 8) {
            const int kk0 = k + half * 2;
            const int kk1 = k + 4 + half * 2;
            v2f a0; a0.x = lrow[kk0]; a0.y = lrow[kk0 + 1];
            v2f b0; b0.x = g[kk0];    b0.y = g[kk0 + 1];
            v2f a1; a1.x = lrow[kk1]; a1.y = lrow[kk1 + 1];
            v2f b1; b1.x = g[kk1];    b1.y = g[kk1 + 1];
            acc0 = __builtin_amdgcn_wmma_f32_16x16x4_f32(
                false, a0, false, b0, (short)0, acc0, false, false);
            acc1 = __builtin_amdgcn_wmma_f32_16x16x4_f32(
                false, a1, false, b1, (short)0, acc1, false, false);
            n2 = fmaf(a0.x, a0.x, n2); n2 = fmaf(a0.y, a0.y, n2);
            n2 = fmaf(a1.x, a1.x, n2); n2 = fmaf(a1.y, a1.y, n2);
        }
        v8f acc = acc0 + acc1;
        // combine the two K-halves of the per-token squared norm
        n2 += __shfl_xor(n2, 16, 32);
        if (half == 0) s_n2[tile * 16 + m] = n2;
        // D layout: VGPR v, lanes 0-15 -> M=v,N=lane ; lanes 16-31 -> M=v+8.
        // All columns equal; lanes with N==0 write 8 rows each.
        if (m == 0) {
            const int base = tile * 16 + half * 8;
            #pragma unroll
            for (int v = 0; v < 8; ++v) s_dot[base + v] = acc[v];
        }
    }
    __syncthreads();

    // ---- distances = dot / (gn * ln) ----
    const float gn = s_gn;
    for (int i = tid; i < TOKENS; i += 256) {
        float ln = fmaxf(sqrtf(s_n2[i]), 1e-8f);
        s_dot[i] = s_dot[i] / (gn * ln);
    }
    __syncthreads();

    // ---- iterative top-8 (ties -> lowest index, matching jax top_k set) ----
    int min_ix = 1 << 30, max_ix = -1, min_iy = 1 << 30, max_iy = -1;
    for (int it = 0; it < NTOP; ++it) {
        float bv = -__builtin_inff();
        int   bi = TOKENS;
        for (int i = tid; i < TOKENS; i += 256) {
            float v = s_dot[i];
            if (v > bv || (v == bv && i < bi)) { bv = v; bi = i; }
        }
        s_red[tid] = bv; s_redi[tid] = bi;
        __syncthreads();
        for (int s = 128; s > 0; s >>= 1) {
            if (tid < s) {
                float ov = s_red[tid + s]; int oi = s_redi[tid + s];
                if (ov > s_red[tid] || (ov == s_red[tid] && oi < s_redi[tid])) {
                    s_red[tid] = ov; s_redi[tid] = oi;
                }
            }
            __syncthreads();
        }
        if (tid == 0) {
            const int idx = s_redi[0];
            const int ix = idx / FH, iy = idx % FH;
            min_ix = min(min_ix, ix); max_ix = max(max_ix, ix);
            min_iy = min(min_iy, iy); max_iy = max(max_iy, iy);
            s_dot[idx] = -__builtin_inff();
        }
        __syncthreads();
    }

    if (tid == 0) {
        const int x_i = min_ix * PATCH, x_f = max_ix * PATCH;
        const int y_i = min_iy * PATCH, y_f = max_iy * PATCH;
        int4 bx;
        bx.x = max(x_i, 0);                              // x0
        bx.y = min(max(x_f, x_i + PATCH), IMG);          // x1
        bx.z = max(y_i, 0);                              // y0
        bx.w = min(max(y_f, y_i + PATCH), IMG);          // y1
        boxes[b] = bx;
    }
}

// ---------------------------------------------------------------------------
// Kernel 2: per-pixel bilinear crop-resize (align_corners=True).
// Memory-bound gather: ~154 MB write + <=154 MB read.
// ---------------------------------------------------------------------------
__global__ void __launch_bounds__(256)
crop_resize_kernel(const float* __restrict__ images,
                   const int4* __restrict__ boxes,
                   float* __restrict__ out)
{
    const int idx = blockIdx.x * 256 + threadIdx.x;   // < 64*3*448*448
    const int tx = idx % IMG;
    const int ty = (idx / IMG) % IMG;
    const int c  = (idx / (IMG * IMG)) % 3;
    const int b  = idx / (IMG * IMG * 3);

    const int4 bx = boxes[b];                 // {x0, x1, y0, y1}
    const float h = (float)(bx.y - bx.x);
    const float w = (float)(bx.w - bx.z);
    const float scale = 1.0f / (float)(IMG - 1);

    const float sy = (float)bx.x + (float)ty * (h - 1.0f) * scale;
    const float sx = (float)bx.z + (float)tx * (w - 1.0f) * scale;
    const int ylo = (int)floorf(sy);
    const int xlo = (int)floorf(sx);
    const int yhi = min(ylo + 1, bx.y - 1);
    const int xhi = min(xlo + 1, bx.w - 1);
    const float wy = sy - (float)ylo;
    const float wx = sx - (float)xlo;

    const float* __restrict__ img = images + ((size_t)b * 3 + c) * IMG * IMG;
    const float v00 = img[ylo * IMG + xlo];
    const float v01 = img[ylo * IMG + xhi];
    const float v10 = img[yhi * IMG + xlo];
    const float v11 = img[yhi * IMG + xhi];

    out[idx] = (1.0f - wy) * (1.0f - wx) * v00
             + (1.0f - wy) * wx          * v01
             + wy          * (1.0f - wx) * v10
             + wy          * wx          * v11;
}

// ---------------------------------------------------------------------------
extern "C" void kernel_launch(void* const* d_in, const int* in_sizes, int n_in,
                              void* d_out, int out_size, void* d_ws, size_t ws_size,
                              hipStream_t stream)
{
    const float* x      = (const float*)d_in[0];   // (64, 785, 768) f32
    const float* images = (const float*)d_in[1];   // (64, 3, 448, 448) f32
    float* out  = (float*)d_out;                   // (64, 3, 448, 448) f32
    int4* boxes = (int4*)d_ws;                     // 64 * int4

    gl_sim_topk_kernel<<<64, 256, 0, stream>>>(x, boxes);

    const int total  = 64 * 3 * IMG * IMG;         // 38,535,168
    const int blocks = total / 256;                // 150,528 exactly
    crop_resize_kernel<<<blocks, 256, 0, stream>>>(images, boxes, out);
}